// KalmanFilter_45131516346848
// MI455X (gfx1250) — compile-verified
//
#include <hip/hip_runtime.h>
#include <hip/hip_bf16.h>

// Problem constants (match reference): B=256 groups, T=1000 steps, S=16 state, M=4 measures
#define NB 256
#define NT 1000
#define NS 16
#define NM 4

typedef __attribute__((ext_vector_type(2))) float v2f;
typedef __attribute__((ext_vector_type(8))) float v8f;

// ---------------------------------------------------------------------------
// Wave-level 16x16x16 fp32 matmuls from 4 chained V_WMMA_F32_16X16X4_F32.
// All matrices 16x16 row-major in LDS. Fragment layouts per ISA 7.12.2 (wave32):
//   A 16x4 : lane L, vgpr v -> row = L&15,           k  = v + 2*(L>>4)
//   B 4x16 : lane L, vgpr v -> k   = v + 2*(L>>4),   col= L&15
//   C 16x16: lane L, vgpr v -> row = v + 8*(L>>4),   col= L&15
// ---------------------------------------------------------------------------
__device__ __forceinline__ v8f wave_mm16_core(const float* A, const float* B,
                                              v8f c, int lo, int hi) {
#pragma unroll
  for (int k0 = 0; k0 < 16; k0 += 4) {
    v2f a, b;
    a.x = A[lo * 16 + k0 + 2 * hi + 0];
    a.y = A[lo * 16 + k0 + 2 * hi + 1];
    b.x = B[(k0 + 2 * hi + 0) * 16 + lo];
    b.y = B[(k0 + 2 * hi + 1) * 16 + lo];
    c = __builtin_amdgcn_wmma_f32_16x16x4_f32(false, a, false, b, (short)0, c,
                                              false, false);
  }
  return c;
}

__device__ __forceinline__ void wave_mm16_z(const float* A, const float* B,
                                            float* C) {
  const int lane = (int)(threadIdx.x & 31);
  const int lo = lane & 15;
  const int hi = lane >> 4;
  v8f c;
#pragma unroll
  for (int r = 0; r < 8; ++r) c[r] = 0.0f;
  c = wave_mm16_core(A, B, c, lo, hi);
#pragma unroll
  for (int r = 0; r < 8; ++r) C[(r + 8 * hi) * 16 + lo] = c[r];
}

__device__ __forceinline__ void wave_mm16_acc(const float* A, const float* B,
                                              float* C, const float* Cin) {
  const int lane = (int)(threadIdx.x & 31);
  const int lo = lane & 15;
  const int hi = lane >> 4;
  v8f c;
#pragma unroll
  for (int r = 0; r < 8; ++r) c[r] = Cin[(r + 8 * hi) * 16 + lo];
  c = wave_mm16_core(A, B, c, lo, hi);
#pragma unroll
  for (int r = 0; r < 8; ++r) C[(r + 8 * hi) * 16 + lo] = c[r];
}

// ---------------------------------------------------------------------------
// Kernel 1: covariance / gain recursion. Batch-independent -> ONE wave does the
// whole T=1000 chain. 7 matmuls = 28 v_wmma per step. Emits cov_p[t] (16x16)
// and K[t] (16x4) to workspace with b128 traffic.
// ---------------------------------------------------------------------------
__global__ void __launch_bounds__(32)
kf_gain_kernel(const float* __restrict__ gF, const float* __restrict__ gH,
               const float* __restrict__ gQ, const float* __restrict__ gR,
               const float* __restrict__ gP0,
               float* __restrict__ covseq, float* __restrict__ kseq) {
  __shared__ __align__(16) float sF[256], sFT[256], sQ[256], sHp[256], sHT[256], sRp[256];
  __shared__ __align__(16) float sCov[256], sTmp[256], sCovp[256], sHC[256], sSm[256];
  __shared__ __align__(16) float sSinv[256], sPHt[256], sK[256], sKH[256];
  const int lane = (int)threadIdx.x;

  // Setup: F, F^T, Q, H padded to 16x16 (rows>=4 zero), H^T padded (cols>=4
  // zero), R padded (top-left 4x4), cov0. Sinv pre-zeroed once (only the 4x4
  // block is rewritten each step; zero padding makes all padded matmuls exact).
  for (int e = lane; e < 256; e += 32) {
    int i = e >> 4, j = e & 15;
    sF[e] = gF[e];
    sFT[e] = gF[j * 16 + i];
    sQ[e] = gQ[e];
    sHp[e] = (i < NM) ? gH[i * 16 + j] : 0.0f;
    sHT[e] = (j < NM) ? gH[j * 16 + i] : 0.0f;
    sRp[e] = (i < NM && j < NM) ? gR[i * NM + j] : 0.0f;
    sCov[e] = gP0[e];
    sSinv[e] = 0.0f;
  }
  __syncthreads();

  for (int t = 0; t < NT; ++t) {
    wave_mm16_z(sF, sCov, sTmp);           // F*P
    wave_mm16_acc(sTmp, sFT, sCovp, sQ);   // cov_p = F*P*F^T + Q

    // stream cov_p[t] out: 2 x b128 per lane
    {
      const float4* cp4 = (const float4*)sCovp;
      float4* gp4 = (float4*)(covseq + (size_t)t * 256);
#pragma unroll
      for (int e4 = lane; e4 < 64; e4 += 32) gp4[e4] = cp4[e4];
    }

    wave_mm16_z(sHp, sCovp, sHC);          // Hcov = H*cov_p  (rows 0..3)
    wave_mm16_acc(sHC, sHT, sSm, sRp);     // S = Hcov*H^T + R (top-left 4x4)

    // 4x4 inverse of SPD innovation covariance: Gauss-Jordan, redundant in
    // every lane (registers only), lanes 0..15 write the result block.
    float m[4][4], inv[4][4];
#pragma unroll
    for (int r = 0; r < 4; ++r)
#pragma unroll
      for (int c2 = 0; c2 < 4; ++c2) {
        m[r][c2] = sSm[r * 16 + c2];
        inv[r][c2] = (r == c2) ? 1.0f : 0.0f;
      }
#pragma unroll
    for (int p = 0; p < 4; ++p) {
      float d = 1.0f / m[p][p];
#pragma unroll
      for (int c2 = 0; c2 < 4; ++c2) { m[p][c2] *= d; inv[p][c2] *= d; }
#pragma unroll
      for (int r = 0; r < 4; ++r) {
        if (r != p) {
          float f = m[r][p];
#pragma unroll
          for (int c2 = 0; c2 < 4; ++c2) {
            m[r][c2] -= f * m[p][c2];
            inv[r][c2] -= f * inv[p][c2];
          }
        }
      }
    }
    if (lane < 16) sSinv[(lane >> 2) * 16 + (lane & 3)] = inv[lane >> 2][lane & 3];
    __syncthreads();

    wave_mm16_z(sCovp, sHT, sPHt);         // P*H^T          (cols 0..3)
    wave_mm16_z(sPHt, sSinv, sK);          // K = P*H^T*S^-1 (cols 0..3)

    // K[t] row-major 16x4: row r = sK[r*16 + 0..3] -> one b128 per lane<16
    if (lane < 16)
      ((float4*)kseq)[t * 16 + lane] = *(const float4*)(sK + lane * 16);

    wave_mm16_z(sK, sHC, sKH);             // K*Hcov (exact: padding is zero)
    // cov_u = symmetrize(cov_p - K*Hcov)
    for (int e = lane; e < 256; e += 32) {
      int i = e >> 4, j = e & 15;
      sCov[e] = sCovp[e] - 0.5f * (sKH[e] + sKH[j * 16 + i]);
    }
    __syncthreads();
  }
}

// ---------------------------------------------------------------------------
// Kernel 2: mean recursion. One wave handles 16 batches as a 16x16 tile
// (rows=batch, cols=state). 3 WMMA matmuls (12 v_wmma) per step; all global
// and LDS side traffic is b128. 16 blocks x 32 threads cover all 256 batches.
// ---------------------------------------------------------------------------
__global__ void __launch_bounds__(32)
kf_mean_kernel(const float* __restrict__ y, const float* __restrict__ gF,
               const float* __restrict__ gH, const float* __restrict__ gMu0,
               const float* __restrict__ kseq, float* __restrict__ means_out) {
  __shared__ __align__(16) float sMt[256], sMp[256], sZ[256], sRe[256];
  __shared__ __align__(16) float sFT[256], sHT[256], sKT[256];
  const int lane = (int)threadIdx.x;
  const int b0 = (int)blockIdx.x * 16;

  for (int e = lane; e < 256; e += 32) {
    int i = e >> 4, j = e & 15;
    (void)i;
    sFT[e] = gF[j * 16 + i];                   // F^T
    sHT[e] = (j < NM) ? gH[j * 16 + i] : 0.0f; // H^T padded (cols>=4 zero)
    sKT[e] = 0.0f;                             // K^T rows>=4 stay zero
    sRe[e] = 0.0f;                             // resid cols>=4 stay zero
    sMt[e] = gMu0[j];                          // mean_u(-1) = init_mean
  }
  __syncthreads();

  for (int t = 0; t < NT; ++t) {
    wave_mm16_z(sMt, sFT, sMp);                // mean_p = mean_u * F^T

    // store mean_p: batch row (16 floats) = 4 float4s -> 2 x b128 per lane
    {
      const float4* mp4 = (const float4*)sMp;
      float4* gm4 = (float4*)means_out;
#pragma unroll
      for (int e4 = lane; e4 < 64; e4 += 32) {
        int bl = e4 >> 2, ch = e4 & 3;
        gm4[(((size_t)(b0 + bl) * NT + t) << 2) + ch] = mp4[e4];
      }
    }

    wave_mm16_z(sMp, sHT, sZ);                 // Z = mean_p * H^T (cols 0..3)

    if (lane < 16) {
      // load K_t row (b128), scatter as K^T column
      float4 kv = ((const float4*)kseq)[t * 16 + lane];
      sKT[0 * 16 + lane] = kv.x;
      sKT[1 * 16 + lane] = kv.y;
      sKT[2 * 16 + lane] = kv.z;
      sKT[3 * 16 + lane] = kv.w;
      // resid = y - Z : one b128 global load + b128 LDS load/store per batch
      float4 yv = ((const float4*)y)[(size_t)(b0 + lane) * NT + t];
      float4 zv = *(const float4*)(sZ + lane * 16);
      float4 rv;
      rv.x = yv.x - zv.x;
      rv.y = yv.y - zv.y;
      rv.z = yv.z - zv.z;
      rv.w = yv.w - zv.w;
      *(float4*)(sRe + lane * 16) = rv;
    }
    __syncthreads();
    wave_mm16_acc(sRe, sKT, sMt, sMp);         // mean_u = mean_p + resid*K^T
    __syncthreads();
  }
}

// ---------------------------------------------------------------------------
// Kernel 3: broadcast cov_p[t] to all 256 batches (262 MB, the HBM-bound part)
// + copy R,H tail. grid = (250, 256), 256 threads; x covers T*256 floats as
// 64000 float4 per batch, y = batch.
// ---------------------------------------------------------------------------
__global__ void __launch_bounds__(256)
kf_bcast_kernel(const float4* __restrict__ covseq4, float4* __restrict__ covs4,
                const float* __restrict__ gR, const float* __restrict__ gH,
                float* __restrict__ tail) {
  const int idx = (int)blockIdx.x * 256 + (int)threadIdx.x; // 0..63999
  float4 v = covseq4[idx];
  covs4[(size_t)blockIdx.y * (NT * NS * NS / 4) + idx] = v;
  if (blockIdx.y == 0 && blockIdx.x == 0) {
    int tid = (int)threadIdx.x;
    if (tid < NM * NM) tail[tid] = gR[tid];
    else if (tid < NM * NM + NM * NS) tail[tid] = gH[tid - NM * NM];
  }
}

extern "C" void kernel_launch(void* const* d_in, const int* in_sizes, int n_in,
                              void* d_out, int out_size, void* d_ws, size_t ws_size,
                              hipStream_t stream) {
  (void)in_sizes; (void)n_in; (void)out_size; (void)ws_size;
  // setup_inputs order: input, F, H, Q, R, init_mean, init_cov (all fp32)
  const float* inp  = (const float*)d_in[0];
  const float* F    = (const float*)d_in[1];
  const float* H    = (const float*)d_in[2];
  const float* Q    = (const float*)d_in[3];
  const float* R    = (const float*)d_in[4];
  const float* mu0  = (const float*)d_in[5];
  const float* P0   = (const float*)d_in[6];

  float* out = (float*)d_out;
  float* means_out = out;                                  // [B,T,S]
  float* covs_out  = out + (size_t)NB * NT * NS;           // [B,T,S,S]
  float* tail      = covs_out + (size_t)NB * NT * NS * NS; // R(16) then H(64)

  float* covseq = (float*)d_ws;               // [T,16,16] = 1,024,000 B
  float* kseq   = covseq + (size_t)NT * 256;  // [T,16,4]  =   256,000 B

  // 1) batch-invariant covariance/gain chain (one wave, 28 wmma/step)
  kf_gain_kernel<<<1, 32, 0, stream>>>(F, H, Q, R, P0, covseq, kseq);
  // 2) data-dependent mean chains, 16 batches per wave (12 wmma/step)
  kf_mean_kernel<<<16, 32, 0, stream>>>(inp, F, H, mu0, kseq, means_out);
  // 3) HBM-rate broadcast of covariances + R,H tail
  dim3 bgrid(NT * NS * NS / 4 / 256, NB); // (250, 256)
  kf_bcast_kernel<<<bgrid, 256, 0, stream>>>((const float4*)covseq,
                                             (float4*)covs_out, R, H, tail);
}